// MPNLayer_79877801771104
// MI455X (gfx1250) — compile-verified
//
#include <hip/hip_runtime.h>
#include <math.h>

typedef float v2f __attribute__((ext_vector_type(2)));
typedef float v4f __attribute__((ext_vector_type(4)));
typedef float v8f __attribute__((ext_vector_type(8)));

#define BATCH  128
#define HIDDEN 1024
#define INPUT  1024
#define ETA    0.01f
#define LAMBDA 0.95f

// ---------------------------------------------------------------------------
// Kernel 1: y0[b,h] = bias[h] + sum_i x[b,i]*W[h,i]   (f32 GEMM via WMMA)
// One wave per 16x16 output tile. 8 batch-tiles x 64 hidden-tiles = 512 waves.
// A = x tile (16x4, f32), B = W^T tile (4x16, f32), C/D f32 16x16.
// Bias folded into accumulator init (each C element gets bias[col] once).
// ---------------------------------------------------------------------------
__global__ __launch_bounds__(256) void mpn_gemm_wmma(
    const float* __restrict__ x, const float* __restrict__ W,
    const float* __restrict__ bias, float* __restrict__ y0)
{
    const int wave  = blockIdx.x * (blockDim.x >> 5) + (threadIdx.x >> 5);
    const int lane  = threadIdx.x & 31;
    const int half  = lane >> 4;     // 0: lanes 0-15, 1: lanes 16-31
    const int idx   = lane & 15;

    const int btile = wave >> 6;     // 0..7   (BATCH/16)
    const int htile = wave & 63;     // 0..63  (HIDDEN/16)
    const int brow  = btile * 16;
    const int hbase = htile * 16;

    const float bv = bias[hbase + idx];    // per-lane column bias
    v8f acc;
#pragma unroll
    for (int v = 0; v < 8; ++v) acc[v] = bv;

    const float* xrow = x + (size_t)(brow  + idx) * INPUT;  // A row (M = idx)
    const float* wrow = W + (size_t)(hbase + idx) * INPUT;  // B col (N = idx)

#pragma unroll 4
    for (int k0 = 0; k0 < INPUT; k0 += 4) {
        const int k = k0 + 2 * half;       // lanes 0-15: K0/K1, lanes 16-31: K2/K3
        v2f a, b;
        a[0] = xrow[k];     a[1] = xrow[k + 1];
        b[0] = wrow[k];     b[1] = wrow[k + 1];
        // D = A(16x4) * B(4x16) + C   — v_wmma_f32_16x16x4_f32
        acc = __builtin_amdgcn_wmma_f32_16x16x4_f32(
                  /*neg_a=*/false, a, /*neg_b=*/false, b,
                  /*c_mod=*/(short)0, acc, /*reuse_a=*/false, /*reuse_b=*/false);
    }

    // C/D layout: VGPR v, lane<16 -> M=v ; lane>=16 -> M=v+8 ; N = idx
#pragma unroll
    for (int v = 0; v < 8; ++v) {
        const int row = brow + v + 8 * half;
        y0[(size_t)row * HIDDEN + hbase + idx] = acc[v];
    }
}

// ---------------------------------------------------------------------------
// Kernel 2: single pass over M (the 1 GB stream -> roofline limiter).
// One 256-thread block per (b,h) row: 4 floats of M per thread held in
// registers across {reduce, tanh, update} so M is read exactly once.
//   ym    = sum_i W[h,i]*M[b,h,i]*x[b,i]
//   hval  = tanh(y0[b,h] + ym)            (y0 staged in d_out h-region)
//   M_new = LAMBDA*M + ETA*hval*x
// Non-temporal on the M stream keeps W/x resident in the 192MB L2.
// ---------------------------------------------------------------------------
__global__ __launch_bounds__(256) void mpn_update(
    const float* __restrict__ x, const float* __restrict__ M,
    const float* __restrict__ W, float* __restrict__ out)
{
    const int bh  = blockIdx.x;        // b*HIDDEN + h
    const int b   = bh >> 10;
    const int hh  = bh & (HIDDEN - 1);
    const int tid = threadIdx.x;
    const int i0  = tid * 4;

    const size_t mOff = ((size_t)bh << 10) + (size_t)i0;
    v4f m  = __builtin_nontemporal_load((const v4f*)(M + mOff));
    v4f w  = *(const v4f*)(W + ((size_t)hh << 10) + i0);
    v4f xv = *(const v4f*)(x + ((size_t)b  << 10) + i0);

    float partial = 0.0f;
#pragma unroll
    for (int j = 0; j < 4; ++j)
        partial += w[j] * m[j] * xv[j];   // modulated part only (x*W done by WMMA)

    // wave32 butterfly reduction
#pragma unroll
    for (int off = 16; off >= 1; off >>= 1)
        partial += __shfl_xor(partial, off, 32);

    __shared__ float red[8];
    __shared__ float hshare;
    if ((tid & 31) == 0) red[tid >> 5] = partial;
    __syncthreads();

    if (tid == 0) {
        float s = 0.0f;
#pragma unroll
        for (int j = 0; j < 8; ++j) s += red[j];
        const float y    = out[bh] + s;      // out[bh] currently holds y0
        const float hval = tanhf(y);
        out[bh] = hval;                      // overwrite y0 with final h
        hshare  = hval;
    }
    __syncthreads();
    const float hval = hshare;

    v4f mn;
    const float ehx = ETA * hval;
#pragma unroll
    for (int j = 0; j < 4; ++j)
        mn[j] = LAMBDA * m[j] + ehx * xv[j];

    __builtin_nontemporal_store(
        mn, (v4f*)(out + (size_t)BATCH * HIDDEN + mOff));
}

extern "C" void kernel_launch(void* const* d_in, const int* in_sizes, int n_in,
                              void* d_out, int out_size, void* d_ws, size_t ws_size,
                              hipStream_t stream)
{
    (void)in_sizes; (void)n_in; (void)out_size; (void)d_ws; (void)ws_size;
    const float* x     = (const float*)d_in[0];   // (128, 1024)
    const float* state = (const float*)d_in[1];   // (128, 1024, 1024)
    const float* W     = (const float*)d_in[2];   // (1024, 1024)
    const float* bias  = (const float*)d_in[3];   // (1024,)
    float* out = (float*)d_out;                   // [h (128*1024) | M_new (128*1024*1024)]

    // Stage y0 = bias + x @ W^T into the h-region of out (WMMA GEMM).
    mpn_gemm_wmma<<<dim3(64), dim3(256), 0, stream>>>(x, W, bias, out);
    // Single-pass M stream: reduce + tanh + Hebbian update.
    mpn_update<<<dim3(BATCH * HIDDEN), dim3(256), 0, stream>>>(x, state, W, out);
}